// FS_MI_80058190397911
// MI455X (gfx1250) — compile-verified
//
#include <hip/hip_runtime.h>

// LIF-style scan: v += x[t]; s = (v > thr[t]); out[t] = s*d[t]; v -= s*h[t]
// T=64, B=32, D=8192. Memory-bound (128 MB traffic, ~5.5us floor @ 23.3 TB/s).
// CDNA5 path: GLOBAL_LOAD_ASYNC_TO_LDS_B128 deep prefetch + s_wait_asynccnt.

constexpr int T       = 64;
constexpr int Bdim    = 32;
constexpr int Ddim    = 8192;
constexpr unsigned BD       = (unsigned)Bdim * (unsigned)Ddim; // 262144 floats / timestep
constexpr unsigned BD_BYTES = BD * 4u;                         // 1 MiB / timestep
constexpr int BLOCK   = 256;
constexpr int STAGES  = 16;            // async pipeline depth (wait imm hardcoded below)

__device__ __forceinline__ void lif_step(float4& v, const float4 xv,
                                         const float h_t, const float d_t,
                                         const float th_t, float4& o) {
    v.x += xv.x; v.y += xv.y; v.z += xv.z; v.w += xv.w;
    o.x = (v.x > th_t) ? d_t : 0.f;  v.x -= (v.x > th_t) ? h_t : 0.f;
    o.y = (v.y > th_t) ? d_t : 0.f;  v.y -= (v.y > th_t) ? h_t : 0.f;
    o.z = (v.z > th_t) ? d_t : 0.f;  v.z -= (v.z > th_t) ? h_t : 0.f;
    o.w = (v.w > th_t) ? d_t : 0.f;  v.w -= (v.w > th_t) ? h_t : 0.f;
}

__global__ __launch_bounds__(BLOCK) void lif_scan_kernel(
    const float* __restrict__ x, const float* __restrict__ h,
    const float* __restrict__ dm, const float* __restrict__ thr,
    float* __restrict__ out)
{
    __shared__ float4 buf[STAGES * BLOCK];           // 64 KB staging
    __shared__ float  sh_h[T], sh_d[T], sh_th[T];

    const int tid = threadIdx.x;
    if (tid < T) { sh_h[tid] = h[tid]; sh_d[tid] = dm[tid]; sh_th[tid] = thr[tid]; }
    __syncthreads();

    // float index of this thread's 4-wide column within one (B,D) slab
    const unsigned elem      = blockIdx.x * (BLOCK * 4u) + (unsigned)tid * 4u;
    const unsigned base_byte = elem * 4u;

    // LDS byte address of this thread's slot (flat addr low 32 bits == LDS addr)
    const unsigned lds_base =
        (unsigned)(unsigned long long)(&buf[0]) + (unsigned)tid * 16u;

    // ---- prologue: fill the async pipeline (per-lane 16B -> LDS) ----
#pragma unroll
    for (int t = 0; t < STAGES; ++t) {
        unsigned voff  = (unsigned)t * BD_BYTES + base_byte;
        unsigned laddr = lds_base + (unsigned)t * (BLOCK * 16u);
        asm volatile("global_load_async_to_lds_b128 %0, %1, %2"
                     :: "v"(laddr), "v"(voff), "s"(x) : "memory");
    }

    float4 v = make_float4(0.f, 0.f, 0.f, 0.f);

    // ---- main loop: consume stage t, keep STAGES loads in flight ----
    // Async loads complete in order, so ASYNCcnt <= STAGES-1 guarantees load t done.
    for (int t = 0; t <= T - STAGES; ++t) {
        asm volatile("s_wait_asynccnt 15" ::: "memory");
        const int stage = t & (STAGES - 1);
        const float4 xv  = buf[stage * BLOCK + tid];
        const float h_t  = sh_h[t], d_t = sh_d[t], th_t = sh_th[t];
        float4 o;
        lif_step(v, xv, h_t, d_t, th_t, o);
        *(float4*)(out + (unsigned)t * BD + elem) = o;

        const int tn = t + STAGES;
        if (tn < T) {
            unsigned voff  = (unsigned)tn * BD_BYTES + base_byte;
            unsigned laddr = lds_base + (unsigned)(tn & (STAGES - 1)) * (BLOCK * 16u);
            asm volatile("global_load_async_to_lds_b128 %0, %1, %2"
                         :: "v"(laddr), "v"(voff), "s"(x) : "memory");
        }
    }

    // ---- tail: everything issued long ago; drain and finish ----
    asm volatile("s_wait_asynccnt 0" ::: "memory");
    for (int t = T - STAGES + 1; t < T; ++t) {
        const int stage = t & (STAGES - 1);
        const float4 xv  = buf[stage * BLOCK + tid];
        const float h_t  = sh_h[t], d_t = sh_d[t], th_t = sh_th[t];
        float4 o;
        lif_step(v, xv, h_t, d_t, th_t, o);
        *(float4*)(out + (unsigned)t * BD + elem) = o;
    }
}

extern "C" void kernel_launch(void* const* d_in, const int* in_sizes, int n_in,
                              void* d_out, int out_size, void* d_ws, size_t ws_size,
                              hipStream_t stream) {
    (void)in_sizes; (void)n_in; (void)d_ws; (void)ws_size; (void)out_size;
    const float* x   = (const float*)d_in[0];   // (T, B, D) f32
    const float* h   = (const float*)d_in[1];   // (T,) f32
    const float* dm  = (const float*)d_in[2];   // (T,) f32
    const float* thr = (const float*)d_in[3];   // (T,) f32
    float* out = (float*)d_out;                 // (T, B, D) f32

    const unsigned grid = BD / (BLOCK * 4u);    // 256 blocks, each owns 1024 floats of state
    lif_scan_kernel<<<dim3(grid), dim3(BLOCK), 0, stream>>>(x, h, dm, thr, out);
}